// HungarianMatcher_24137716203567
// MI455X (gfx1250) — compile-verified
//
#include <hip/hip_runtime.h>

typedef __attribute__((ext_vector_type(2))) float v2f;
typedef __attribute__((ext_vector_type(8))) float v8f;

#define NOBJ 16
#define NNOUN 8
#define DIM  512
#define EPSM 1e-6f
#define BIGM 1e30f

// One wave (32 lanes) per batch element.
//  Phase 1: cost[16][8] = -(O.N)/(|O||N|+eps) via 128x V_WMMA_F32_16X16X4_F32 (K=512).
//           Noun rows are clamped (not zero-padded): WMMA columns 8..15 of D are
//           garbage-but-finite and never read, so no per-iteration predication is needed.
//  Phase 2: wave-parallel Jonker-Volgenant on cost^T (8 rows x 16 cols, virtual col 16).
__global__ __launch_bounds__(32) void HungarianMatcher_kernel(
    const float* __restrict__ objects,   // [B,16,512]
    const float* __restrict__ nouns,     // [B, 8,512]
    float* __restrict__ out,             // [B*16] indices (as float) then [B*128] cost
    int B)
{
    __shared__ float cost_l[NOBJ * NNOUN];  // cost[o][n]
    __shared__ float u_l[16];               // dual u[0..8] used

    const int b    = blockIdx.x;
    const int lane = threadIdx.x;      // 0..31
    const int half = lane >> 4;        // K split across wave halves
    const int mrow = lane & 15;        // A: object row M ; B(valid part): noun col N
    const int nrow = mrow & 7;         // clamped noun row for padded lanes

    // v2f element k-stride: per K-step t of 4, lane covers cols 4t+2*half+{0,1}
    const v2f* aP = (const v2f*)objects + (size_t)b * (NOBJ * DIM / 2) + mrow * (DIM / 2) + half;
    const v2f* bP = (const v2f*)nouns   + (size_t)b * (NNOUN * DIM / 2) + nrow * (DIM / 2) + half;
    const bool bvalid = (mrow < NNOUN);

    v8f  acc0 = {};
    v8f  acc1 = {};
    float sqa = 0.f, sqb = 0.f;

#pragma unroll 8
    for (int t = 0; t < DIM / 4; t += 2) {
        v2f a0 = aP[2 * t];
        v2f b0 = bP[2 * t];
        v2f a1 = aP[2 * t + 2];
        v2f b1 = bP[2 * t + 2];
        sqa += a0.x * a0.x + a0.y * a0.y + a1.x * a1.x + a1.y * a1.y;
        sqb += b0.x * b0.x + b0.y * b0.y + b1.x * b1.x + b1.y * b1.y;
        // Two independent accumulation chains keep back-to-back WMMAs unstalled.
        acc0 = __builtin_amdgcn_wmma_f32_16x16x4_f32(false, a0, false, b0,
                                                     (short)0, acc0, false, false);
        acc1 = __builtin_amdgcn_wmma_f32_16x16x4_f32(false, a1, false, b1,
                                                     (short)0, acc1, false, false);
    }
    v8f acc = acc0 + acc1;

    // Pair-reduce squared norms across wave halves (each half saw half of K).
    float on2 = sqa + __shfl_xor(sqa, 16);   // lane L: |obj[L&15]|^2
    float nn2 = sqb + __shfl_xor(sqb, 16);   // lane L (L&15<8): |noun[L&15]|^2
    float nnorm = sqrtf(nn2);

    // Apply cosine normalization; D layout: acc[r] at lane L -> o = r + 8*half, n = L&15.
    float* costOut = out + (size_t)B * NOBJ + (size_t)b * (NOBJ * NNOUN);
#pragma unroll
    for (int r = 0; r < 8; ++r) {
        int   o     = r + 8 * half;
        float onorm = sqrtf(__shfl(on2, o));      // bpermute broadcast of object norm
        float c     = -acc[r] / (onorm * nnorm + EPSM);
        if (bvalid) {
            cost_l[o * NNOUN + mrow]  = c;
            costOut[o * NNOUN + mrow] = c;
        }
    }
    if (lane < 9) u_l[lane] = 0.f;
    __syncthreads();

    // ---- Hungarian on cost^T: n=8 rows, m=16 cols, virtual start column 16 ----
    float vj = 0.f;   // lane j: v[j]
    int   pj = -1;    // lane j: p[j] (column -> assigned row)

    for (int i = 0; i < NNOUN; ++i) {
        if (lane == 16) pj = i;           // p[m] = i
        float minv = BIGM;
        int   way  = 0;
        bool  used = false;
        int   j0   = 16;

        for (int iter = 0; iter < 32; ++iter) {
            int i0 = __shfl(pj, j0);      // uniform: row assigned to current column
            if (i0 == -1) break;
            if (lane == j0) used = true;

            float cur = BIGM;
            if (lane < 16) cur = cost_l[lane * NNOUN + i0] - u_l[i0] - vj;
            bool better = (lane < 16) && !used && (cur < minv);
            if (better) { minv = cur; way = j0; }

            float masked = (!used && lane <= 16) ? minv : BIGM;
            // argmin over lanes with first-index tie-break (matches jnp.argmin)
            float bv = masked; int bi = lane;
#pragma unroll
            for (int off = 16; off > 0; off >>= 1) {
                float ov = __shfl_xor(bv, off);
                int   oi = __shfl_xor(bi, off);
                if (ov < bv || (ov == bv && oi < bi)) { bv = ov; bi = oi; }
            }
            float delta = bv;

            // dual updates: u[p[j]] += delta on used cols (distinct rows); v/minv vector ops
            if (used) atomicAdd(&u_l[pj], delta);
            if (used) vj -= delta; else minv -= delta;
            __syncthreads();              // order ds_add before next u_l read (uniform path)
            j0 = bi;
        }

        // augmenting chain back to the virtual column
        for (int iter = 0; iter < 32 && j0 != 16; ++iter) {
            int j1  = __shfl(way, j0);
            int pj1 = __shfl(pj, j1);
            if (lane == j0) pj = pj1;
            j0 = j1;
        }
    }

    // Emit sorted matched (object, noun) index pairs: rank by ballot-popcount.
    unsigned long long mask = __ballot(lane < 16 && pj != -1);
    if (lane < 16 && pj != -1) {
        int rank = __popcll(mask & ((1ull << lane) - 1ull));
        out[(size_t)b * 16 + rank]     = (float)lane;  // object index
        out[(size_t)b * 16 + 8 + rank] = (float)pj;    // noun index
    }
}

extern "C" void kernel_launch(void* const* d_in, const int* in_sizes, int n_in,
                              void* d_out, int out_size, void* d_ws, size_t ws_size,
                              hipStream_t stream) {
    const float* objects = (const float*)d_in[0];   // [B,16,512] f32
    const float* nouns   = (const float*)d_in[1];   // [B, 8,512] f32
    float* out = (float*)d_out;                     // [B*16] indices then [B*128] cost
    int B = in_sizes[0] / (NOBJ * DIM);
    HungarianMatcher_kernel<<<B, 32, 0, stream>>>(objects, nouns, out, B);
}